// SpatialArgmax2d_15058155340192
// MI455X (gfx1250) — compile-verified
//
#include <hip/hip_runtime.h>
#include <hip/hip_bf16.h>
#include <math.h>

// Problem constants from the reference: (B,N,H,W) = (32,64,128,128), fp32.
#define MAP_W   128
#define MAP_H   128
#define MAP_HW  (MAP_W * MAP_H)      // 16384 floats = 64 KiB per map
#define BLOCK   256                  // 8 wave32 waves per workgroup
#define ITERS   (MAP_HW / 4 / BLOCK) // 16 float4 loads per thread

// One workgroup per (b,n) heatmap. Streaming argmax reduction + parabolic
// sub-pixel refinement done by thread 0 (neighbors are L2-hot).
__global__ __launch_bounds__(BLOCK) void spatial_argmax2d_kernel(
    const float* __restrict__ in, float* __restrict__ out)
{
    const int g = blockIdx.x;                       // map id in [0, B*N)
    const float*  map  = in + (size_t)g * MAP_HW;
    const float4* map4 = (const float4*)map;
    const int t = threadIdx.x;

    float bestV = -INFINITY;
    int   bestI = 0;

    // Each thread scans 16 float4's with stride BLOCK (coalesced b128 loads).
    // Indices ascend with i, so strict '>' keeps the first occurrence.
    #pragma unroll
    for (int i = 0; i < ITERS; ++i) {
        const int p4 = i * BLOCK + t;
        // gfx1250 prefetch path (global_prefetch_b8), 8 iterations ahead.
        if (i < ITERS / 2) {
            __builtin_prefetch((const void*)&map4[p4 + (ITERS / 2) * BLOCK], 0, 0);
        }
        const float4 v = map4[p4];
        const int base = p4 * 4;
        if (v.x > bestV) { bestV = v.x; bestI = base;     }
        if (v.y > bestV) { bestV = v.y; bestI = base + 1; }
        if (v.z > bestV) { bestV = v.z; bestI = base + 2; }
        if (v.w > bestV) { bestV = v.w; bestI = base + 3; }
    }

    // wave32 butterfly reduction: (max value, min index on ties) — matches
    // jnp.argmax first-occurrence semantics across interleaved lane indices.
    #pragma unroll
    for (int off = 16; off > 0; off >>= 1) {
        const float oV = __shfl_xor(bestV, off, 32);
        const int   oI = __shfl_xor(bestI, off, 32);
        if (oV > bestV || (oV == bestV && oI < bestI)) { bestV = oV; bestI = oI; }
    }

    // Cross-wave reduction through LDS (8 waves per block).
    __shared__ float sV[BLOCK / 32];
    __shared__ int   sI[BLOCK / 32];
    const int wave = t >> 5;
    const int lane = t & 31;
    if (lane == 0) { sV[wave] = bestV; sI[wave] = bestI; }
    __syncthreads();

    if (t == 0) {
        float bV = sV[0];
        int   bI = sI[0];
        #pragma unroll
        for (int w = 1; w < BLOCK / 32; ++w) {
            if (sV[w] > bV || (sV[w] == bV && sI[w] < bI)) { bV = sV[w]; bI = sI[w]; }
        }

        const int y = bI >> 7;        // / MAP_W
        const int x = bI & (MAP_W - 1);

        // Replicate ('edge') padding: out-of-range neighbor == center value.
        const float c = map[bI];
        const float l = (x > 0)         ? map[bI - 1]     : c;
        const float r = (x < MAP_W - 1) ? map[bI + 1]     : c;
        const float u = (y > 0)         ? map[bI - MAP_W] : c;
        const float d = (y < MAP_H - 1) ? map[bI + MAP_W] : c;

        const float denx = l - 2.0f * c + r;
        const float deny = u - 2.0f * c + d;
        const float dx = (denx != 0.0f) ? 0.5f * (l - r) / denx : 0.0f;
        const float dy = (deny != 0.0f) ? 0.5f * (u - d) / deny : 0.0f;

        out[2 * g + 0] = (float)x + dx;   // output order is (x, y)
        out[2 * g + 1] = (float)y + dy;
    }
}

extern "C" void kernel_launch(void* const* d_in, const int* in_sizes, int n_in,
                              void* d_out, int out_size, void* d_ws, size_t ws_size,
                              hipStream_t stream) {
    (void)n_in; (void)d_ws; (void)ws_size; (void)out_size;
    const float* in  = (const float*)d_in[0];
    float*       out = (float*)d_out;
    const int nmaps = in_sizes[0] / MAP_HW;   // B*N = 2048
    spatial_argmax2d_kernel<<<nmaps, BLOCK, 0, stream>>>(in, out);
}